// LocalEncoder_36043365548696
// MI455X (gfx1250) — compile-verified
//
#include <hip/hip_runtime.h>
#include <cstdint>

#define Bn 32
#define Nn 1024
#define Sn 512
#define KLOC 32
#define KLAYER 20
#define EPSV 1e-5f
#define FINF 3.402823466e38f

static constexpr int R1 = Bn * Sn * KLOC;    // 524288 rows (pn layers)
static constexpr int RG = Bn * Sn * KLAYER;  // 327680 rows (intra layers)
static constexpr int RS = Bn * Sn;           // 16384 rows (per-point layers)
static constexpr int OUT_MAIN = Bn * 128 * Sn; // 2097152 floats, then new_xyz tail

typedef __attribute__((ext_vector_type(2))) float v2f;
typedef __attribute__((ext_vector_type(8))) float v8f;

// ---------------------------------------------------------------------------
// Farthest point sampling: one workgroup (1024 threads = 32 waves) per batch.
// ---------------------------------------------------------------------------
__global__ void fps_kernel(const float* __restrict__ xyz,
                           float* __restrict__ nxyz,
                           float* __restrict__ out_tail) {
  int b = blockIdx.x;
  int t = threadIdx.x;                 // 0..1023, one point per thread
  const float* Pp = xyz + (size_t)b * Nn * 3;
  float px = Pp[t * 3 + 0], py = Pp[t * 3 + 1], pz = Pp[t * 3 + 2];
  float dist = FINF;
  __shared__ float sval[Nn];
  __shared__ int sidx[Nn];
  __shared__ float lastp[3];
  int last = 0;
  for (int k = 0; k < Sn; ++k) {
    if (t == 0) {
      float lx = Pp[last * 3], ly = Pp[last * 3 + 1], lz = Pp[last * 3 + 2];
      size_t o = ((size_t)b * Sn + k) * 3;
      nxyz[o] = lx; nxyz[o + 1] = ly; nxyz[o + 2] = lz;
      out_tail[o] = lx; out_tail[o + 1] = ly; out_tail[o + 2] = lz;
      lastp[0] = lx; lastp[1] = ly; lastp[2] = lz;
    }
    __syncthreads();
    float dx = px - lastp[0], dy = py - lastp[1], dz = pz - lastp[2];
    dist = fminf(dist, dx * dx + dy * dy + dz * dz);
    sval[t] = dist; sidx[t] = t;
    __syncthreads();
    for (int off = Nn / 2; off > 0; off >>= 1) {
      if (t < off) {
        float bv = sval[t + off]; int bi = sidx[t + off];
        if (bv > sval[t] || (bv == sval[t] && bi < sidx[t])) { sval[t] = bv; sidx[t] = bi; }
      }
      __syncthreads();
    }
    last = sidx[0];   // argmax, first-occurrence tie-break (matches jnp.argmax)
  }
}

// ---------------------------------------------------------------------------
// kNN: one workgroup (256 threads) per query; same distance formula as the
// reference; iterative min-extraction with smallest-index tie-break.
// ---------------------------------------------------------------------------
__global__ void knn_kernel(const float* __restrict__ q, const float* __restrict__ p,
                           int* __restrict__ out, int Np, int Sq, int k) {
  int query = blockIdx.x;              // b*Sq + s
  int b = query / Sq;
  const float* Q = q + (size_t)query * 3;
  const float* P = p + (size_t)b * Np * 3;
  __shared__ float d[1024];
  __shared__ float rv[256];
  __shared__ int ri[256];
  float qx = Q[0], qy = Q[1], qz = Q[2];
  float qq = qx * qx + qy * qy + qz * qz;
  for (int i = threadIdx.x; i < Np; i += blockDim.x) {
    float px = P[i * 3], py = P[i * 3 + 1], pz = P[i * 3 + 2];
    float pp = px * px + py * py + pz * pz;
    d[i] = qq + pp - 2.f * (qx * px + qy * py + qz * pz);
  }
  __syncthreads();
  for (int kk = 0; kk < k; ++kk) {
    float best = FINF; int bi = 0x7fffffff;
    for (int i = threadIdx.x; i < Np; i += blockDim.x) {
      float v = d[i];
      if (v < best || (v == best && i < bi)) { best = v; bi = i; }
    }
    rv[threadIdx.x] = best; ri[threadIdx.x] = bi;
    __syncthreads();
    for (int off = 128; off > 0; off >>= 1) {
      if ((int)threadIdx.x < off) {
        float bv = rv[threadIdx.x + off]; int bj = ri[threadIdx.x + off];
        if (bv < rv[threadIdx.x] || (bv == rv[threadIdx.x] && bj < ri[threadIdx.x])) {
          rv[threadIdx.x] = bv; ri[threadIdx.x] = bj;
        }
      }
      __syncthreads();
    }
    if (threadIdx.x == 0) { out[(size_t)query * k + kk] = ri[0]; d[ri[0]] = FINF; }
    __syncthreads();
  }
}

// ---------------------------------------------------------------------------
// M = gather(xyz, idx) - new_xyz, stored with stride 4 (zero-padded K=3->4)
// so the pn1 GEMM can take the unguarded fast path.
// ---------------------------------------------------------------------------
__global__ void build_M_kernel(const float* __restrict__ xyz, const float* __restrict__ nxyz,
                               const int* __restrict__ idx, float* __restrict__ M4) {
  int i = blockIdx.x * blockDim.x + threadIdx.x;
  if (i >= R1) return;
  int bs = i / KLOC; int b = bs / Sn;
  int j = idx[i];
  const float* P = xyz + ((size_t)b * Nn + j) * 3;
  const float* Cp = nxyz + (size_t)bs * 3;
  M4[(size_t)i * 4 + 0] = P[0] - Cp[0];
  M4[(size_t)i * 4 + 1] = P[1] - Cp[1];
  M4[(size_t)i * 4 + 2] = P[2] - Cp[2];
  M4[(size_t)i * 4 + 3] = 0.f;
}

// Zero-pad pn_w1 (64x3) -> (64x4)
__global__ void pad_w1_kernel(const float* __restrict__ w, float* __restrict__ wp) {
  int i = blockIdx.x * blockDim.x + threadIdx.x;
  if (i >= 64 * 4) return;
  int n = i >> 2, k = i & 3;
  wp[i] = (k < 3) ? w[n * 3 + k] : 0.f;
}

// Zero-pad e*_w3 (3x32) -> (16x32): zero weight rows => exactly-zero padded
// output channels, so the fast N=16 path is numerically identical.
__global__ void pad_w3_kernel(const float* __restrict__ w, float* __restrict__ wp) {
  int i = blockIdx.x * blockDim.x + threadIdx.x;
  if (i >= 16 * 32) return;
  int n = i >> 5, k = i & 31;
  wp[i] = (n < 3) ? w[n * 32 + k] : 0.f;
}

// ---------------------------------------------------------------------------
// Fast WMMA fp32 GEMM: C[M,N] = A[M,K] @ W[N,K]^T, requires K%4==0, N==NT*16.
// One wave owns one 16-row M-tile and ALL NT column tiles (A-fragment reuse):
// inner loop = 1 A b64 load + NT W b64 loads + NT v_wmma, zero branches.
// Fragment indexing per ISA 7.12.2 (16x16x4 f32): lane L -> m/n = L&15,
// k-pair = k0 + 2*(L>>4); C/D row r+8*(L>>4), col L&15.
// ---------------------------------------------------------------------------
template <int NT>
__global__ __launch_bounds__(256) void gemm_wmma_fast(const float* __restrict__ A,
                                                      const float* __restrict__ W,
                                                      float* __restrict__ Cc,
                                                      int Mrows, int Kdim) {
  constexpr int Ncols = NT * 16;
  int wave = (blockIdx.x * blockDim.x + threadIdx.x) >> 5;
  int lane = threadIdx.x & 31;
  if (wave >= (Mrows >> 4)) return;
  int lm = lane & 15;
  int hi = lane >> 4;
  const float* Arow = A + (size_t)((wave << 4) + lm) * Kdim + (hi << 1);
  const float* Wcol = W + (size_t)lm * Kdim + (hi << 1);
  v8f acc[NT];
#pragma unroll
  for (int nt = 0; nt < NT; ++nt) acc[nt] = (v8f){0.f, 0.f, 0.f, 0.f, 0.f, 0.f, 0.f, 0.f};
  for (int k0 = 0; k0 < Kdim; k0 += 4) {
    v2f a = *(const v2f*)(Arow + k0);           // 8B aligned: K,k0,2*hi all even
#pragma unroll
    for (int nt = 0; nt < NT; ++nt) {
      v2f b = *(const v2f*)(Wcol + (size_t)nt * 16 * Kdim + k0);
      acc[nt] = __builtin_amdgcn_wmma_f32_16x16x4_f32(false, a, false, b,
                                                      (short)0, acc[nt], false, false);
    }
  }
  size_t rowbase = (size_t)(wave << 4) * Ncols;
#pragma unroll
  for (int nt = 0; nt < NT; ++nt) {
#pragma unroll
    for (int r = 0; r < 8; ++r) {
      Cc[rowbase + (size_t)(r + (hi << 3)) * Ncols + nt * 16 + lm] = acc[nt][r];
    }
  }
}

// ---------------------------------------------------------------------------
// Deterministic BatchNorm: fixed-partition partials -> finalize -> apply+ReLU.
// ---------------------------------------------------------------------------
#define BN_CHUNK 8192
__global__ void bn_partial_kernel(const float* __restrict__ X, float* __restrict__ psum,
                                  float* __restrict__ psq, int rows, int Ccols, int nchunks) {
  int c = blockIdx.x;
  int chunk = blockIdx.y;
  int r0 = chunk * BN_CHUNK;
  int r1 = min(rows, r0 + BN_CHUNK);
  float s = 0.f, qv = 0.f;
  for (int r = r0 + threadIdx.x; r < r1; r += blockDim.x) {
    float v = X[(size_t)r * Ccols + c];
    s += v; qv += v * v;
  }
  __shared__ float ls[256], lq[256];
  ls[threadIdx.x] = s; lq[threadIdx.x] = qv;
  __syncthreads();
  for (int off = 128; off > 0; off >>= 1) {
    if ((int)threadIdx.x < off) {
      ls[threadIdx.x] += ls[threadIdx.x + off];
      lq[threadIdx.x] += lq[threadIdx.x + off];
    }
    __syncthreads();
  }
  if (threadIdx.x == 0) {
    psum[(size_t)c * nchunks + chunk] = ls[0];
    psq[(size_t)c * nchunks + chunk] = lq[0];
  }
}

__global__ void bn_finalize_kernel(const float* __restrict__ psum, const float* __restrict__ psq,
                                   const float* __restrict__ g, const float* __restrict__ bta,
                                   float* __restrict__ scale, float* __restrict__ shift,
                                   int Ccols, int nchunks, float inv_n) {
  int c = blockIdx.x * blockDim.x + threadIdx.x;
  if (c >= Ccols) return;
  float s = 0.f, qv = 0.f;
  for (int i = 0; i < nchunks; ++i) {
    s += psum[(size_t)c * nchunks + i];
    qv += psq[(size_t)c * nchunks + i];
  }
  float mean = s * inv_n;
  float var = fmaxf(qv * inv_n - mean * mean, 0.f);
  float sc = g[c] * rsqrtf(var + EPSV);
  scale[c] = sc;
  shift[c] = bta[c] - mean * sc;
}

__global__ void bn_apply_relu_kernel(float* __restrict__ X, const float* __restrict__ scale,
                                     const float* __restrict__ shift, int rows, int Ccols) {
  size_t i = (size_t)blockIdx.x * blockDim.x + threadIdx.x;
  size_t tot = (size_t)rows * Ccols;
  if (i >= tot) return;
  int c = (int)(i % Ccols);
  X[i] = fmaxf(X[i] * scale[c] + shift[c], 0.f);
}

// ---------------------------------------------------------------------------
// Segment max-pool over contiguous groups of `seg` rows.
// ---------------------------------------------------------------------------
__global__ void maxpool_seg_kernel(const float* __restrict__ X, float* __restrict__ Y,
                                   int groups, int seg, int Ccols) {
  size_t i = (size_t)blockIdx.x * blockDim.x + threadIdx.x;
  if (i >= (size_t)groups * Ccols) return;
  int c = (int)(i % Ccols);
  int gidx = (int)(i / Ccols);
  const float* row = X + (size_t)gidx * seg * Ccols + c;
  float m = -FINF;
  for (int j = 0; j < seg; ++j) m = fmaxf(m, row[(size_t)j * Ccols]);
  Y[i] = m;
}

// ---------------------------------------------------------------------------
// feature = [inv(128), l2n(e1 dir)(3), l2n(e2 dir)(3)]; e buffers stride 16.
// ---------------------------------------------------------------------------
__global__ void feature_assemble_kernel(const float* __restrict__ inv, const float* __restrict__ e1,
                                        const float* __restrict__ e2, float* __restrict__ feat) {
  int i = blockIdx.x * blockDim.x + threadIdx.x;
  int tot = RS * 134;
  if (i >= tot) return;
  int row = i / 134, c = i % 134;
  float v;
  if (c < 128) {
    v = inv[(size_t)row * 128 + c];
  } else if (c < 131) {
    const float* d = e1 + (size_t)row * 16;
    float nrm = sqrtf(d[0] * d[0] + d[1] * d[1] + d[2] * d[2]);
    v = d[c - 128] / fmaxf(nrm, 1e-12f);
  } else {
    const float* d = e2 + (size_t)row * 16;
    float nrm = sqrtf(d[0] * d[0] + d[1] * d[1] + d[2] * d[2]);
    v = d[c - 131] / fmaxf(nrm, 1e-12f);
  }
  feat[i] = v;
}

// ---------------------------------------------------------------------------
// graph feature: feature part -> c1 input cols 0..255; geometric 16 -> g16.
// ---------------------------------------------------------------------------
__global__ void graph_f_kernel(const float* __restrict__ feat, const int* __restrict__ idx,
                               float* __restrict__ A288) {
  int i = blockIdx.x * blockDim.x + threadIdx.x;
  int tot = RG * 128;
  if (i >= tot) return;
  int c = i & 127;
  int row = i >> 7;
  int bs = row / KLAYER; int b = bs / Sn;
  int j = idx[row];
  float fc = feat[(size_t)bs * 134 + c];
  float nf = feat[((size_t)b * Sn + j) * 134 + c];
  float* out = A288 + (size_t)row * 288;
  out[c] = nf - fc;
  out[128 + c] = fc;
}

__device__ __forceinline__ void cross_l2n(const float* a, const float* b, float* o) {
  o[0] = a[1] * b[2] - a[2] * b[1];
  o[1] = a[2] * b[0] - a[0] * b[2];
  o[2] = a[0] * b[1] - a[1] * b[0];
  float n = sqrtf(o[0] * o[0] + o[1] * o[1] + o[2] * o[2]);
  float inv = 1.f / fmaxf(n, 1e-12f);
  o[0] *= inv; o[1] *= inv; o[2] *= inv;
}

__global__ void graph_geo_kernel(const float* __restrict__ feat, const float* __restrict__ nxyz,
                                 const int* __restrict__ idx, float* __restrict__ g16) {
  int row = blockIdx.x * blockDim.x + threadIdx.x;
  if (row >= RG) return;
  int bs = row / KLAYER; int b = bs / Sn;
  int j = idx[row];
  int nbs = b * Sn + j;
  const float* fC = feat + (size_t)bs * 134;
  const float* fN = feat + (size_t)nbs * 134;
  float R[3][3], Q[3][3];
  R[0][0] = fC[128]; R[0][1] = fC[129]; R[0][2] = fC[130];
  R[1][0] = fC[131]; R[1][1] = fC[132]; R[1][2] = fC[133];
  cross_l2n(R[0], R[1], R[2]);
  Q[0][0] = fN[128]; Q[0][1] = fN[129]; Q[0][2] = fN[130];
  Q[1][0] = fN[131]; Q[1][1] = fN[132]; Q[1][2] = fN[133];
  cross_l2n(Q[0], Q[1], Q[2]);
  const float* pc = nxyz + (size_t)bs * 3;
  const float* pn = nxyz + (size_t)nbs * 3;
  float rel[3] = {pn[0] - pc[0], pn[1] - pc[1], pn[2] - pc[2]};
  float* o = g16 + (size_t)row * 16;
  for (int i2 = 0; i2 < 3; ++i2)
    for (int j2 = 0; j2 < 3; ++j2)
      o[i2 * 3 + j2] = R[i2][0] * Q[j2][0] + R[i2][1] * Q[j2][1] + R[i2][2] * Q[j2][2];
  for (int i2 = 0; i2 < 3; ++i2)
    o[9 + i2] = rel[0] * R[i2][0] + rel[1] * R[i2][1] + rel[2] * R[i2][2];
  for (int i2 = 0; i2 < 3; ++i2)
    o[12 + i2] = rel[0] * Q[i2][0] + rel[1] * Q[i2][1] + rel[2] * Q[i2][2];
  o[15] = sqrtf(rel[0] * rel[0] + rel[1] * rel[1] + rel[2] * rel[2]);
}

__global__ void pos_copy_kernel(const float* __restrict__ pos, float* __restrict__ A288) {
  int i = blockIdx.x * blockDim.x + threadIdx.x;
  int tot = RG * 32;
  if (i >= tot) return;
  int c = i & 31;
  int row = i >> 5;
  A288[(size_t)row * 288 + 256 + c] = pos[i];
}

__global__ void out_transpose_kernel(const float* __restrict__ pooled, float* __restrict__ out) {
  int i = blockIdx.x * blockDim.x + threadIdx.x;
  if (i >= OUT_MAIN) return;
  int s = i % Sn;
  int c = (i / Sn) % 128;
  int b = i / (Sn * 128);
  out[i] = pooled[((size_t)b * Sn + s) * 128 + c];
}

// ---------------------------------------------------------------------------
// Host-side helpers (stream-ordered, graph-capture safe)
// ---------------------------------------------------------------------------
static inline int cdiv(long long a, long long b) { return (int)((a + b - 1) / b); }

static void run_gemm_fast(hipStream_t stream, const float* A, const float* W, float* C,
                          int Mrows, int Kdim, int Ncols) {
  int blocks = cdiv((long long)(Mrows / 16) * 32, 256);
  if (Ncols == 16)
    gemm_wmma_fast<1><<<blocks, 256, 0, stream>>>(A, W, C, Mrows, Kdim);
  else if (Ncols == 32)
    gemm_wmma_fast<2><<<blocks, 256, 0, stream>>>(A, W, C, Mrows, Kdim);
  else if (Ncols == 64)
    gemm_wmma_fast<4><<<blocks, 256, 0, stream>>>(A, W, C, Mrows, Kdim);
  else
    gemm_wmma_fast<8><<<blocks, 256, 0, stream>>>(A, W, C, Mrows, Kdim);
}

static void run_bn_relu(hipStream_t stream, float* X, int rows, int Ccols,
                        const float* g, const float* b, float* psum, float* psq,
                        float* scale, float* shift) {
  int nchunks = (rows + BN_CHUNK - 1) / BN_CHUNK;
  dim3 grid(Ccols, nchunks);
  bn_partial_kernel<<<grid, 256, 0, stream>>>(X, psum, psq, rows, Ccols, nchunks);
  bn_finalize_kernel<<<1, 128, 0, stream>>>(psum, psq, g, b, scale, shift, Ccols, nchunks,
                                            1.0f / (float)rows);
  bn_apply_relu_kernel<<<cdiv((long long)rows * Ccols, 256), 256, 0, stream>>>(X, scale, shift,
                                                                               rows, Ccols);
}

extern "C" void kernel_launch(void* const* d_in, const int* in_sizes, int n_in,
                              void* d_out, int out_size, void* d_ws, size_t ws_size,
                              hipStream_t stream) {
  (void)in_sizes; (void)out_size; (void)ws_size;
  const float* xyz = (const float*)d_in[0];
  const float* P[42];
  for (int i = 0; i < 42 && i < n_in; ++i) P[i] = (const float*)d_in[i];
  const float *pn_w1 = P[1],  *pn_g1 = P[2],  *pn_b1 = P[3];
  const float *pn_w2 = P[4],  *pn_g2 = P[5],  *pn_b2 = P[6];
  const float *pn_w3 = P[7],  *pn_g3 = P[8],  *pn_b3 = P[9];
  const float *inv_w1 = P[10], *inv_g1 = P[11], *inv_b1 = P[12];
  const float *inv_w2 = P[13], *inv_g2 = P[14], *inv_b2 = P[15];
  const float *e1_w1 = P[16], *e1_g1 = P[17], *e1_b1 = P[18];
  const float *e1_w2 = P[19], *e1_g2 = P[20], *e1_b2 = P[21];
  const float *e1_w3 = P[22];
  const float *e2_w1 = P[23], *e2_g1 = P[24], *e2_b1 = P[25];
  const float *e2_w2 = P[26], *e2_g2 = P[27], *e2_b2 = P[28];
  const float *e2_w3 = P[29];
  const float *pos_w = P[30], *pos_g = P[31], *pos_b = P[32];
  const float *c1_w = P[33], *c1_g = P[34], *c1_b = P[35];
  const float *c2_w = P[36], *c2_g = P[37], *c2_b = P[38];
  const float *c3_w = P[39], *c3_g = P[40], *c3_b = P[41];

  float* base = (float*)d_ws;
  size_t off = 0;
  auto alloc = [&](size_t n) -> float* {
    float* p = base + off;
    off += (n + 63) & ~(size_t)63;
    return p;
  };
  float* newxyz = alloc((size_t)RS * 3);
  int* idx_loc  = (int*)alloc((size_t)R1);
  int* idx_g    = (int*)alloc((size_t)RG);
  float* psum   = alloc(128 * 64);
  float* psq    = alloc(128 * 64);
  float* scale  = alloc(128);
  float* shift  = alloc(128);
  float* w1pad  = alloc(64 * 4);
  float* w3pad1 = alloc(16 * 32);
  float* w3pad2 = alloc(16 * 32);
  float* M4buf  = alloc((size_t)R1 * 4);
  float* xi     = alloc((size_t)RS * 128);
  float* tA     = alloc((size_t)RS * 128);
  float* tB     = alloc((size_t)RS * 128);
  float* eT1    = alloc((size_t)RS * 64);
  float* eT2    = alloc((size_t)RS * 32);
  float* eD1    = alloc((size_t)RS * 16);
  float* eD2    = alloc((size_t)RS * 16);
  float* feat   = alloc((size_t)RS * 134);
  float* g16    = alloc((size_t)RG * 16);
  float* posb   = alloc((size_t)RG * 32);
  float* pooled = alloc((size_t)RS * 128);
  float* BufA   = alloc((size_t)RG * 288);   // also holds (R1 x 64/128) activations
  float* BufB   = alloc((size_t)R1 * 128);

  float* out = (float*)d_out;

  // --- Phase 1: FPS + local kNN + local frames -----------------------------
  fps_kernel<<<Bn, Nn, 0, stream>>>(xyz, newxyz, out + OUT_MAIN);
  knn_kernel<<<Bn * Sn, 256, 0, stream>>>(newxyz, xyz, idx_loc, Nn, Sn, KLOC);
  build_M_kernel<<<cdiv(R1, 256), 256, 0, stream>>>(xyz, newxyz, idx_loc, M4buf);
  pad_w1_kernel<<<1, 256, 0, stream>>>(pn_w1, w1pad);
  pad_w3_kernel<<<2, 256, 0, stream>>>(e1_w3, w3pad1);
  pad_w3_kernel<<<2, 256, 0, stream>>>(e2_w3, w3pad2);

  // --- Phase 2: disnet (PointNet MLP + invariant/equivariant heads) --------
  run_gemm_fast(stream, M4buf, w1pad, BufA, R1, 4, 64);
  run_bn_relu(stream, BufA, R1, 64, pn_g1, pn_b1, psum, psq, scale, shift);
  run_gemm_fast(stream, BufA, pn_w2, BufB, R1, 64, 128);
  run_bn_relu(stream, BufB, R1, 128, pn_g2, pn_b2, psum, psq, scale, shift);
  run_gemm_fast(stream, BufB, pn_w3, BufA, R1, 128, 128);
  run_bn_relu(stream, BufA, R1, 128, pn_g3, pn_b3, psum, psq, scale, shift);
  maxpool_seg_kernel<<<cdiv((long long)RS * 128, 256), 256, 0, stream>>>(BufA, xi, RS, KLOC, 128);

  run_gemm_fast(stream, xi, inv_w1, tA, RS, 128, 128);
  run_bn_relu(stream, tA, RS, 128, inv_g1, inv_b1, psum, psq, scale, shift);
  run_gemm_fast(stream, tA, inv_w2, tB, RS, 128, 128);
  run_bn_relu(stream, tB, RS, 128, inv_g2, inv_b2, psum, psq, scale, shift);

  run_gemm_fast(stream, xi, e1_w1, eT1, RS, 128, 64);
  run_bn_relu(stream, eT1, RS, 64, e1_g1, e1_b1, psum, psq, scale, shift);
  run_gemm_fast(stream, eT1, e1_w2, eT2, RS, 64, 32);
  run_bn_relu(stream, eT2, RS, 32, e1_g2, e1_b2, psum, psq, scale, shift);
  run_gemm_fast(stream, eT2, w3pad1, eD1, RS, 32, 16);

  run_gemm_fast(stream, xi, e2_w1, eT1, RS, 128, 64);
  run_bn_relu(stream, eT1, RS, 64, e2_g1, e2_b1, psum, psq, scale, shift);
  run_gemm_fast(stream, eT1, e2_w2, eT2, RS, 64, 32);
  run_bn_relu(stream, eT2, RS, 32, e2_g2, e2_b2, psum, psq, scale, shift);
  run_gemm_fast(stream, eT2, w3pad2, eD2, RS, 32, 16);

  feature_assemble_kernel<<<cdiv(RS * 134, 256), 256, 0, stream>>>(tB, eD1, eD2, feat);

  // --- Phase 3: graph feature + intra MLP ----------------------------------
  knn_kernel<<<Bn * Sn, 256, 0, stream>>>(newxyz, newxyz, idx_g, Sn, Sn, KLAYER);
  graph_f_kernel<<<cdiv((long long)RG * 128, 256), 256, 0, stream>>>(feat, idx_g, BufA);
  graph_geo_kernel<<<cdiv(RG, 256), 256, 0, stream>>>(feat, newxyz, idx_g, g16);

  run_gemm_fast(stream, g16, pos_w, posb, RG, 16, 32);
  run_bn_relu(stream, posb, RG, 32, pos_g, pos_b, psum, psq, scale, shift);
  pos_copy_kernel<<<cdiv(RG * 32, 256), 256, 0, stream>>>(posb, BufA);

  run_gemm_fast(stream, BufA, c1_w, BufB, RG, 288, 128);
  run_bn_relu(stream, BufB, RG, 128, c1_g, c1_b, psum, psq, scale, shift);
  run_gemm_fast(stream, BufB, c2_w, BufA, RG, 128, 128);
  run_bn_relu(stream, BufA, RG, 128, c2_g, c2_b, psum, psq, scale, shift);
  run_gemm_fast(stream, BufA, c3_w, BufB, RG, 128, 128);
  run_bn_relu(stream, BufB, RG, 128, c3_g, c3_b, psum, psq, scale, shift);

  maxpool_seg_kernel<<<cdiv((long long)RS * 128, 256), 256, 0, stream>>>(BufB, pooled, RS, KLAYER, 128);
  out_transpose_kernel<<<cdiv(OUT_MAIN, 256), 256, 0, stream>>>(pooled, out);
}